// NCACrossEntropy_24352464569138
// MI455X (gfx1250) — compile-verified
//
#include <hip/hip_runtime.h>
#include <hip/hip_bf16.h>
#include <stdint.h>

// NCA cross-entropy, B=512, N=131072 f32. Pure HBM streaming: 256 MB @ 23.3 TB/s => ~11us floor.
// TDM (tensor_load_to_lds) stages 16x512 f32 tiles into LDS (double buffered, padded stride
// for conflict-free banks); WMMA f32 16x16x4 with B=ones is the per-16-row accumulation engine.

#define NCA_B 512
#define NCA_N 131072
#define ROWS_PER_TILE 16
#define ROW_TILES     (NCA_B / ROWS_PER_TILE)              // 32
#define WAVES_PER_BLOCK 8
#define CHUNK_COLS    512                                   // cols staged per TDM transfer
#define CHUNKS_PER_BLOCK 8                                  // 8 * 512 = 4096 cols per WG
#define COLS_PER_BLOCK (CHUNK_COLS * CHUNKS_PER_BLOCK)      // 4096
#define COL_GROUPS    (NCA_N / COLS_PER_BLOCK)              // 32
#define CHUNKS_PER_ROW (COL_GROUPS * WAVES_PER_BLOCK)       // 256 partials per row
#define COLS_PER_WAVE_CHUNK (CHUNK_COLS / WAVES_PER_BLOCK)  // 64

// LDS tile: 512 dwords/row + 2dw pad every 256dw => row stride 516 dwords = 2064 B
#define LDS_ROW_STRIDE 2064
#define LDS_BUF_BYTES  (ROWS_PER_TILE * LDS_ROW_STRIDE)     // 33024
#define LDS_TOTAL      (2 * LDS_BUF_BYTES)                  // 66048 (double buffer)

typedef float v2f  __attribute__((ext_vector_type(2)));
typedef float v8f  __attribute__((ext_vector_type(8)));
typedef unsigned int u32x4 __attribute__((ext_vector_type(4)));
typedef int   i32x4 __attribute__((ext_vector_type(4)));
typedef int   i32x8 __attribute__((ext_vector_type(8)));

#if __has_builtin(__builtin_amdgcn_tensor_load_to_lds)
#define USE_TDM 1
#else
#define USE_TDM 0
#warning "gfx1250: __builtin_amdgcn_tensor_load_to_lds unavailable; using cooperative-copy staging"
#endif

#if !__has_builtin(__builtin_amdgcn_wmma_f32_16x16x4_f32)
#warning "gfx1250: __builtin_amdgcn_wmma_f32_16x16x4_f32 not available"
#endif

__device__ __forceinline__ v8f wmma_rowsum_acc(v2f a, v8f acc) {
#if __has_builtin(__builtin_amdgcn_wmma_f32_16x16x4_f32)
    const v2f ones = {1.0f, 1.0f};
    // D[m,n] = sum_k A[m,k]*1 + C[m,n] -> row sums replicated across n
    return __builtin_amdgcn_wmma_f32_16x16x4_f32(false, a, false, ones,
                                                 (short)0, acc, false, false);
#else
    acc[0] += a[0] + a[1];
    return acc;
#endif
}

#if USE_TDM
// Issue one TDM 2D tile load: 16 rows x 512 cols of f32, tensor row stride N,
// LDS padding: +2 dwords after every 256 dwords (row stride 516 dw, bank-conflict free).
__device__ __forceinline__ void tdm_issue(const float* gsrc, uint32_t ldsAddr) {
    const uint64_t ga = (uint64_t)(uintptr_t)gsrc;
    u32x4 g0;
    g0[0] = 1u;                                     // count=1 (valid), user mode, no gather
    g0[1] = ldsAddr;                                // lds_addr [63:32]
    g0[2] = (uint32_t)ga;                           // global_addr [95:64]
    g0[3] = (uint32_t)((ga >> 32) & 0x01FFFFFFu)    // global_addr [120:96]
          | 0x80000000u;                            // type=2 ("image") [127:126]
    i32x8 g1;
    g1[0] = (2 << 16)        // data_size = 4 bytes
          | (1 << 20)        // pad_enable
          | (7 << 22)        // pad_interval: 256 dwords
          | (1 << 25);       // pad_amount: 2 dwords
    g1[1] = 0;                                       // atomic barrier addr / tensor_dim0 lo16=0
    g1[2] = (int)((NCA_N >> 16) | (ROWS_PER_TILE << 16)); // tensor_dim0 hi16, tensor_dim1 lo16
    g1[3] = (int)(CHUNK_COLS << 16);                 // tensor_dim1 hi16=0, tile_dim0=512
    g1[4] = ROWS_PER_TILE;                           // tile_dim1=16, tile_dim2=0
    g1[5] = NCA_N;                                   // tensor_dim0_stride lo32
    g1[6] = 0;                                       // stride hi16, tensor_dim1_stride lo16
    g1[7] = 0;                                       // tensor_dim1_stride hi32
    const i32x4 z4 = {0, 0, 0, 0};
#if defined(__clang_major__) && (__clang_major__ >= 23)
    const i32x8 z8 = {0, 0, 0, 0, 0, 0, 0, 0};
    __builtin_amdgcn_tensor_load_to_lds(g0, g1, z4, z4, z8, 0);
#else
    __builtin_amdgcn_tensor_load_to_lds(g0, g1, z4, z4, 0);
#endif
}
#endif

__global__ __launch_bounds__(256)
void nca_partial_sums(const float* __restrict__ x,
                      const int*   __restrict__ labels,
                      const int*   __restrict__ indexes,
                      float2*      __restrict__ part)   // [NCA_B][CHUNKS_PER_ROW]
{
    __shared__ __align__(16) char smem[LDS_TOTAL];

    const int lane     = threadIdx.x & 31;
    const int wave     = threadIdx.x >> 5;
    const int rowTile  = blockIdx.y;                 // 0..31
    const int colGroup = blockIdx.x;                 // 0..31
    const int rowBase  = rowTile * ROWS_PER_TILE;
    const int rowM     = lane & 15;                  // A-layout M index for this lane
    const int row      = rowBase + rowM;
    const int half     = lane >> 4;                  // 0 -> K=0,1 ; 1 -> K=2,3
    const int blockColBase = colGroup * COLS_PER_BLOCK;

    const int   yRow = labels[indexes[row]];
    const uint32_t ldsBase = (uint32_t)(uintptr_t)(&smem[0]);  // LDS flat addr: low 32 bits = offset

    v8f accAll  = {0.f,0.f,0.f,0.f,0.f,0.f,0.f,0.f};
    v8f accSame = {0.f,0.f,0.f,0.f,0.f,0.f,0.f,0.f};

    // ---- staging helpers -------------------------------------------------
    auto stage = [&](int c) {
        const int buf = c & 1;
#if USE_TDM
        if (wave == 0) {
            tdm_issue(x + (size_t)rowBase * NCA_N + blockColBase + c * CHUNK_COLS,
                      ldsBase + (uint32_t)(buf * LDS_BUF_BYTES));
        }
#else
        // cooperative copy, coalesced float4, same padded layout as TDM
        char* dst = smem + buf * LDS_BUF_BYTES;
        for (int idx = threadIdx.x; idx < ROWS_PER_TILE * (CHUNK_COLS / 4); idx += 256) {
            const int r  = idx >> 7;                 // 128 float4 per row
            const int c4 = idx & 127;
            const int cc = c4 * 4;
            const float4 v = *(const float4*)(x + (size_t)(rowBase + r) * NCA_N
                                                + blockColBase + c * CHUNK_COLS + cc);
            *(float4*)(dst + r * LDS_ROW_STRIDE + cc * 4 + ((cc >= 256) ? 8 : 0)) = v;
        }
#endif
    };

    // ---- double-buffered pipeline ---------------------------------------
    stage(0);
    const int waveCol = wave * COLS_PER_WAVE_CHUNK;          // this wave's cols in chunk
    const int padShift = (waveCol >= 256) ? 8 : 0;           // uniform per wave
    const char* ldsRow0 = smem;                              // buffer selected per chunk

    for (int c = 0; c < CHUNKS_PER_BLOCK; ++c) {
#if USE_TDM
        if (wave == 0) __builtin_amdgcn_s_wait_tensorcnt(0); // chunk c resident in LDS
#endif
        __syncthreads();                                     // data visible + prev buffer free
        if (c + 1 < CHUNKS_PER_BLOCK) stage(c + 1);          // overlap next DMA with compute

        const char* buf = ldsRow0 + (c & 1) * LDS_BUF_BYTES
                        + rowM * LDS_ROW_STRIDE
                        + (waveCol + 2 * half) * 4 + padShift;
        const int chunkColBase = blockColBase + c * CHUNK_COLS + waveCol + 2 * half;

        #pragma unroll 4
        for (int k = 0; k < COLS_PER_WAVE_CHUNK; k += 4) {
            const float2 xv = *(const float2*)(buf + 4 * k);   // ds_load_b64, conflict-free
            const int col = chunkColBase + k;
            const int l0 = labels[col];
            const int l1 = labels[col + 1];
            const float e0 = __expf(xv.x);
            const float e1 = __expf(xv.y);
            v2f aAll  = { e0, e1 };
            v2f aSame = { (l0 == yRow) ? e0 : 0.0f,
                          (l1 == yRow) ? e1 : 0.0f };
            accAll  = wmma_rowsum_acc(aAll,  accAll);
            accSame = wmma_rowsum_acc(aSame, accSame);
        }
    }

    // D layout (16x16 f32 C/D): lanes 0-15 comp v = row v; lanes 16-31 comp v = row v+8.
    const int chunkIdx = colGroup * WAVES_PER_BLOCK + wave;
    if (lane == 0) {
        #pragma unroll
        for (int m = 0; m < 8; ++m)
            part[(size_t)(rowBase + m) * CHUNKS_PER_ROW + chunkIdx] =
                make_float2(accAll[m], accSame[m]);
    } else if (lane == 16) {
        #pragma unroll
        for (int m = 0; m < 8; ++m)
            part[(size_t)(rowBase + 8 + m) * CHUNKS_PER_ROW + chunkIdx] =
                make_float2(accAll[m], accSame[m]);
    }
}

__global__ __launch_bounds__(NCA_B)
void nca_finalize(const float* __restrict__ x,
                  const int*   __restrict__ indexes,
                  const float2* __restrict__ part,
                  float* __restrict__ out)
{
    __shared__ float red[NCA_B];
    const int row = threadIdx.x;

    float sumAll = 0.0f, sumSame = 0.0f;
    const float2* __restrict__ p = part + (size_t)row * CHUNKS_PER_ROW;
    #pragma unroll 8
    for (int c = 0; c < CHUNKS_PER_ROW; ++c) {       // fixed order -> deterministic
        const float2 v = p[c];
        sumAll  += v.x;
        sumSame += v.y;
    }

    const int   idx   = indexes[row];
    const float eself = __expf(x[(size_t)row * (size_t)NCA_N + idx]);
    const float Z     = sumAll  - eself;             // exclude self column
    const float ps    = sumSame - eself;             // self always same-label
    const float prob  = ps / Z;
    red[row] = (prob != 0.0f) ? __logf(prob) : 0.0f;
    __syncthreads();

    #pragma unroll
    for (int s = NCA_B / 2; s > 0; s >>= 1) {        // deterministic tree reduce
        if (row < s) red[row] += red[row + s];
        __syncthreads();
    }
    if (row == 0) out[0] = -red[0] / (float)NCA_B;
}

extern "C" void kernel_launch(void* const* d_in, const int* in_sizes, int n_in,
                              void* d_out, int out_size, void* d_ws, size_t ws_size,
                              hipStream_t stream) {
    (void)in_sizes; (void)n_in; (void)out_size; (void)ws_size;
    const float* x       = (const float*)d_in[0];   // [512][131072] f32
    const int*   labels  = (const int*)d_in[1];     // [131072]
    const int*   indexes = (const int*)d_in[2];     // [512]
    float*       out     = (float*)d_out;           // [1]
    float2*      part    = (float2*)d_ws;           // 512*256*8B = 1 MB scratch

    dim3 grid(COL_GROUPS, ROW_TILES);               // 32 x 32 = 1024 WGs, 8 waves each
    nca_partial_sums<<<grid, WAVES_PER_BLOCK * 32, 0, stream>>>(x, labels, indexes, part);
    nca_finalize<<<1, NCA_B, 0, stream>>>(x, indexes, part, out);
}